// PieceWisePlanarRegularization_3D_ERP_10393820857123
// MI455X (gfx1250) — compile-verified
//
#include <hip/hip_runtime.h>

// Problem constants (match the reference).
#define HH 512
#define WW 1024
#define NPIX (HH * WW)          // 524288
#define KNB 15
#define GAMMA_F 0.1f
#define TPB 256                 // 8 wave32 waves per block
#define NBLK_MAIN 1024

typedef __attribute__((ext_vector_type(2))) float v2f;
typedef __attribute__((ext_vector_type(8))) float v8f;

// ---------------------------------------------------------------------------
// Exact 32-lane sum via V_WMMA_F32_16X16X4_F32 (wave32).
// A-matrix 16x4 layout (ISA 7.12.2): VGPR0 holds K=0 (lanes 0-15) / K=2
// (lanes 16-31); VGPR1 holds K=1/K=3 (we set those to 0). With B = ones,
// D[m][n] = A[m][0] + A[m][2] = x[m] + x[m+16] for every n.
// D layout: VGPR r, lane l<16 -> D[r][l]; lane l>=16 -> D[r+8][l-16].
// Summing the 8 D VGPRs per lane gives x[0..7]+x[16..23] on lanes 0-15 and
// x[8..15]+x[24..31] on lanes 16-31; one shfl_xor(16) completes the reduction.
// ---------------------------------------------------------------------------
__device__ __forceinline__ float wave_sum_wmma(float x) {
    v2f a; a[0] = x;    a[1] = 0.0f;
    v2f b; b[0] = 1.0f; b[1] = 1.0f;
    v8f c = {0.f, 0.f, 0.f, 0.f, 0.f, 0.f, 0.f, 0.f};
    v8f d = __builtin_amdgcn_wmma_f32_16x16x4_f32(
        /*neg_a=*/false, a, /*neg_b=*/false, b,
        /*c_mod=*/(short)0, c, /*reuse_a=*/false, /*reuse_b=*/false);
    float t = d[0] + d[1] + d[2] + d[3] + d[4] + d[5] + d[6] + d[7];
    t += __shfl_xor(t, 16, 32);
    return t;   // full 32-lane sum in every lane
}

__device__ __forceinline__ float block_sum(float x, float* sm) {
    float wsum = wave_sum_wmma(x);      // EXEC is all-1s here (no divergence)
    int lane = threadIdx.x & 31;
    int wid  = threadIdx.x >> 5;
    if (lane == 0) sm[wid] = wsum;
    __syncthreads();
    float s = 0.0f;
    if (threadIdx.x == 0) {
        #pragma unroll
        for (int i = 0; i < TPB / 32; ++i) s += sm[i];
    }
    return s;   // valid on thread 0 only
}

// ---------------------------------------------------------------------------
// Pass 1: pack p = sig1 * dirs (+ pn = p.n) and n into float4 SoA in L2-
// resident workspace so the random neighbour gathers become 2x b128 loads.
// ---------------------------------------------------------------------------
__global__ void __launch_bounds__(TPB)
precompute_kernel(const float* __restrict__ sig1,
                  const float* __restrict__ sig2,
                  float4* __restrict__ P,       // (px,py,pz,pn)
                  float4* __restrict__ Q) {     // (nx,ny,nz,0)
    int j = blockIdx.x * blockDim.x + threadIdx.x;
    if (j >= NPIX) return;
    int row = j >> 10;          // W == 1024
    int col = j & (WW - 1);
    const float PIf = 3.14159265358979323846f;
    float lat = (0.5f - (row + 0.5f) * (1.0f / (float)HH)) * PIf;
    float lon = ((col + 0.5f) * (1.0f / (float)WW) - 0.5f) * (2.0f * PIf);
    float cl = cosf(lat), sl = sinf(lat);
    float so = sinf(lon), co = cosf(lon);
    float s1 = sig1[j];
    float px = s1 * cl * so;
    float py = s1 * sl;
    float pz = s1 * cl * co;
    float nx = sig2[j], ny = sig2[NPIX + j], nz = sig2[2 * NPIX + j];
    float pn = px * nx + py * ny + pz * nz;
    P[j] = make_float4(px, py, pz, pn);
    Q[j] = make_float4(nx, ny, nz, 0.0f);
}

// ---------------------------------------------------------------------------
// Pass 2: the gather/reduce. Memory-bound: ~115 MB DRAM traffic total
// (neighbours + weights streamed; p/n gathers hit the 192 MB L2).
// nb2 views the int64 neighbour array as int32 pairs; indices < 2^19, so the
// low (little-endian) dword is the full value — 4 B instead of 8 B per index
// in registers.
// ---------------------------------------------------------------------------
__global__ void __launch_bounds__(TPB)
main_kernel(const float4* __restrict__ P,
            const float4* __restrict__ Q,
            const float* __restrict__ weights,
            const int* __restrict__ nb2,
            float* __restrict__ partials) {
    const float scale = 1.0f / (float)NPIX;   // MULTIPLIER == 1.0
    float acc = 0.0f;
    for (int j = blockIdx.x * blockDim.x + threadIdx.x; j < NPIX;
         j += gridDim.x * blockDim.x) {
        float4 pj = P[j];
        float4 qj = Q[j];
        float pn = pj.w;
        float nx = qj.x, ny = qj.y, nz = qj.z;
        float t1 = 0.0f, t2 = 0.0f;
        #pragma unroll
        for (int k = 0; k < KNB; ++k) {
            size_t e = (size_t)k * (size_t)NPIX + (size_t)j;
            int idx  = nb2[2 * e];          // low dword of the int64 index
            float wk = weights[e];
            float4 pb = P[idx];             // global_load_b128 (L2 hit)
            float4 qb = Q[idx];             // global_load_b128 (L2 hit)
            float a1 = pn - (pb.x * nx + pb.y * ny + pb.z * nz);
            float u  = wk * a1;
            t1 += u * u;
            float ex = nx - qb.x, ey = ny - qb.y, ez = nz - qb.z;
            t2 += wk * sqrtf(ex * ex + ey * ey + ez * ez);
        }
        acc += sqrtf(t1) + GAMMA_F * t2;
    }
    acc *= scale;

    __shared__ float sm[TPB / 32];
    float s = block_sum(acc, sm);
    if (threadIdx.x == 0) partials[blockIdx.x] = s;
}

// ---------------------------------------------------------------------------
// Pass 3: deterministic finalize (fixed summation order; no atomics).
// ---------------------------------------------------------------------------
__global__ void __launch_bounds__(TPB)
finalize_kernel(const float* __restrict__ partials, int nPartials,
                float* __restrict__ out) {
    float acc = 0.0f;
    for (int i = threadIdx.x; i < nPartials; i += TPB) acc += partials[i];
    __shared__ float sm[TPB / 32];
    float s = block_sum(acc, sm);
    if (threadIdx.x == 0) out[0] = s;
}

extern "C" void kernel_launch(void* const* d_in, const int* in_sizes, int n_in,
                              void* d_out, int out_size, void* d_ws, size_t ws_size,
                              hipStream_t stream) {
    const float* sig1    = (const float*)d_in[0];   // (1,1,H,W) f32
    const float* sig2    = (const float*)d_in[1];   // (1,3,H,W) f32
    const float* weights = (const float*)d_in[2];   // (K,N)     f32
    const int*   nb2     = (const int*)d_in[3];     // (K,N) int64 viewed as int pairs

    float4* P        = (float4*)d_ws;               // N * 16 B
    float4* Q        = P + NPIX;                    // N * 16 B
    float*  partials = (float*)(Q + NPIX);          // NBLK_MAIN * 4 B

    precompute_kernel<<<NPIX / TPB, TPB, 0, stream>>>(sig1, sig2, P, Q);
    main_kernel<<<NBLK_MAIN, TPB, 0, stream>>>(P, Q, weights, nb2, partials);
    finalize_kernel<<<1, TPB, 0, stream>>>(partials, NBLK_MAIN, (float*)d_out);
}